// SDFNetwork_15874199126510
// MI455X (gfx1250) — compile-verified
//
#include <hip/hip_runtime.h>
#include <cstdint>

#define RES      96
#define CCH      129
#define NPTS     524288
#define SCALE_INV (1.0f / 1.5f)
#define PT_TILES 512
#define STRIP    (NPTS / PT_TILES)   // 1024 points per block
#define BLOCK    256
#define CPB      16                  // channels per block (y grid dim)
#define PPI      (BLOCK / CPB)       // 16 points per loop iteration
#define ITERS    (STRIP / PPI)       // 64

// Byte offset of a statically allocated LDS object (addrspace(3) base is 0).
__device__ __forceinline__ unsigned lds_byte_off(const void* p) {
  return (unsigned)(unsigned long long)(__attribute__((address_space(3))) const void*)p;
}

__global__ __launch_bounds__(BLOCK) void grid_sample_kernel(
    const float* __restrict__ xin, const float* __restrict__ grid,
    float* __restrict__ out) {
  __shared__ float shc[STRIP * 3];   // 12 KB coordinate stage

  const int tid      = threadIdx.x;
  const int tileBase = blockIdx.x * STRIP;
  const int cb       = blockIdx.y;

  // ---- CDNA5 async-to-LDS staging of the coordinate strip (3072 floats) ----
  {
    const float*   src   = xin + (size_t)tileBase * 3;   // 16B-aligned (12288B tiles)
    const unsigned lbase = lds_byte_off(shc);
#pragma unroll
    for (int k = 0; k < 3; ++k) {
      const unsigned idx  = (unsigned)tid + (unsigned)k * BLOCK;  // 768 float4 total
      const unsigned ldst = lbase + idx * 16u;                    // LDS byte address
      const unsigned goff = idx * 16u;                            // 32-bit voffset (GVS)
      asm volatile("global_load_async_to_lds_b128 %0, %1, %2"
                   :: "v"(ldst), "v"(goff), "s"(src)
                   : "memory");
    }
    asm volatile("s_wait_asynccnt 0x0" ::: "memory");
  }
  __syncthreads();

  const int  cl   = tid & (CPB - 1);
  const int  pl   = tid >> 4;             // wave32: 2 points x 16 channels per wave
  const int  c    = cb * CPB + cl;
  const bool cval = (c < CCH);
  const float* gc = grid + (size_t)c * (RES * RES * RES);  // 3.54 MB slab per channel

  // Unroll x2: 16 independent random gathers in flight per thread -> better
  // latency hiding for the HBM/L2-bound corner reads.
#pragma unroll 2
  for (int it = 0; it < ITERS; ++it) {
    const int   lp = it * PPI + pl;
    const float xr = shc[3 * lp + 0];
    const float yr = shc[3 * lp + 1];
    const float zr = shc[3 * lp + 2];
    const bool mask = (fabsf(xr) < 1.5f) && (fabsf(yr) < 1.5f) && (fabsf(zr) < 1.5f);

    // p = (clip(x/1.5,-1,1)+1)*0.5*(R-1), align_corners=True
    const float gx = fminf(fmaxf(xr * SCALE_INV, -1.0f), 1.0f);
    const float gy = fminf(fmaxf(yr * SCALE_INV, -1.0f), 1.0f);
    const float gz = fminf(fmaxf(zr * SCALE_INV, -1.0f), 1.0f);
    const float fx = (gx + 1.0f) * (0.5f * (RES - 1));
    const float fy = (gy + 1.0f) * (0.5f * (RES - 1));
    const float fz = (gz + 1.0f) * (0.5f * (RES - 1));
    // clamp-to-face via i0=min(floor,R-2); w hits exactly 1.0 at the face,
    // which reproduces the reference's clip(i0+1) semantics with no OOB reads.
    const int ix = min((int)floorf(fx), RES - 2);
    const int iy = min((int)floorf(fy), RES - 2);
    const int iz = min((int)floorf(fz), RES - 2);
    const float wx = fx - (float)ix;
    const float wy = fy - (float)iy;
    const float wz = fz - (float)iz;

    if (cval) {
      // 32-bit element offset within the channel slab (slab = 884736 elems)
      const unsigned off = ((unsigned)iz * RES + (unsigned)iy) * RES + (unsigned)ix;
      const float* r00 = gc + off;
      const float* r01 = r00 + RES;
      const float* r10 = r00 + RES * RES;
      const float* r11 = r10 + RES;
      const float a00 = r00[0] + wx * (r00[1] - r00[0]);
      const float a01 = r01[0] + wx * (r01[1] - r01[0]);
      const float a10 = r10[0] + wx * (r10[1] - r10[0]);
      const float a11 = r11[0] + wx * (r11[1] - r11[0]);
      const float b0  = a00 + wy * (a01 - a00);
      const float b1  = a10 + wy * (a11 - a10);
      float v = b0 + wz * (b1 - b0);
      v = mask ? v : 0.0f;
      const int n = tileBase + lp;
      // output is streamed once: keep it out of L2 so the grid slab stays hot
      __builtin_nontemporal_store(v, out + (size_t)n * CCH + c);
    }
  }
}

extern "C" void kernel_launch(void* const* d_in, const int* in_sizes, int n_in,
                              void* d_out, int out_size, void* d_ws, size_t ws_size,
                              hipStream_t stream) {
  const float* x    = (const float*)d_in[0];   // (N,3) float32
  const float* grid = (const float*)d_in[1];   // (1,C,R,R,R) float32
  float*       out  = (float*)d_out;           // (N,C) float32

  dim3 g(PT_TILES, (CCH + CPB - 1) / CPB);     // (512, 9): y = channel slab (L2-blocked)
  grid_sample_kernel<<<g, BLOCK, 0, stream>>>(x, grid, out);
}